// Resampling_79353815760907
// MI455X (gfx1250) — compile-verified
//
#include <hip/hip_runtime.h>

// Problem constants (B, P, H, W, D, C) = (16, 4, 32, 32, 32, 8)
#define NB 16
#define NP 4
#define NH 32
#define NW 32
#define ND 32
#define NC 8
#define NVOX (NB * NP * NH * NW * ND)          // 2,097,152 voxels
#define SLAB (NH * NW * ND * NC)               // 262,144 floats per (b,p)

typedef __attribute__((ext_vector_type(2))) float v2f;
typedef __attribute__((ext_vector_type(8))) float v8f;

// One voxel per lane. Each wave32 covers 32 consecutive voxels: (b,p,h,w) fixed,
// d = lane. Coordinates come from V_WMMA_F32_16X16X4_F32:
//   D(16x16) = A(16x4: theta rows 0..2, zero-padded) x B(4x16: [w,h,d,1] columns) + C(+2.0 rows 0..2)
// D layout: row M in VGPR M, column N in lane N -> lane n gets voxel n's (x,y,z)
// in d[0],d[1],d[2]. Two WMMAs cover columns 0..15 and 16..31; a shuffle moves
// the second result into lanes 16..31.
__global__ __launch_bounds__(256) void Resampling_79353815760907_kernel(
    const float* __restrict__ in,      // (B,P,H,W,D,C)
    const float* __restrict__ theta,   // (B,P,3,4)
    float* __restrict__ out)           // (B,P,H,W,D,C)
{
    const int lane  = threadIdx.x & 31;
    const int row   = lane & 15;
    const int wave  = threadIdx.x >> 5;
    const int vbase = blockIdx.x * 256 + (wave << 5);   // first voxel of this wave
    const int vid   = vbase + lane;                     // this lane's voxel
    const int bp    = vbase >> 15;                      // (b*P + p), uniform per block

    // Uniform-per-wave spatial position of the 32-voxel run (d varies by lane).
    const float wf = (float)((vbase >> 5)  & 31);
    const float hf = (float)((vbase >> 10) & 31);

    // ---- A operand: theta, M=0..2 valid, K=4. 32-bit 16x4 layout:
    //      lanes 0-15 hold K={0,1} in VGPR{0,1}; lanes 16-31 hold K={2,3}.
    const float* tb = theta + bp * 12;
    const int kb = (lane < 16) ? 0 : 2;
    v2f a;
    a[0] = (row < 3) ? tb[row * 4 + kb + 0] : 0.0f;
    a[1] = (row < 3) ? tb[row * 4 + kb + 1] : 0.0f;

    // ---- B operands: columns n = voxel (vbase [+16] + n) as [w, h, d, 1].
    //      lanes 0-15 supply rows K={0,1} = {w,h}; lanes 16-31 rows K={2,3} = {d,1}.
    const bool lo16 = (lane < 16);
    v2f b1, b2;
    b1[0] = lo16 ? wf : (float)row;
    b1[1] = lo16 ? hf : 1.0f;
    b2[0] = lo16 ? wf : (float)(row + 16);
    b2[1] = lo16 ? hf : 1.0f;

    // ---- C operand: +2.0 on rows 0..2 (the reference's pad-coordinate shift).
    v8f c = {};
    c[0] = 2.0f; c[1] = 2.0f; c[2] = 2.0f;

    v8f dA = __builtin_amdgcn_wmma_f32_16x16x4_f32(false, a, false, b1, (short)0, c, false, false);
    v8f dB = __builtin_amdgcn_wmma_f32_16x16x4_f32(false, a, false, b2, (short)0, c, false, false);

    // Columns 16..31 live in lanes 0..15 of dB; shuffle them into lanes 16..31.
    const float xb = __shfl(dB[0], row, 32);
    const float yb = __shfl(dB[1], row, 32);
    const float zb = __shfl(dB[2], row, 32);
    const float X = lo16 ? dA[0] : xb;   // already includes +2.0
    const float Y = lo16 ? dA[1] : yb;
    const float Z = lo16 ? dA[2] : zb;

    // ---- Reference semantics: clip(floor(coord), 0, H+2=34) on ALL axes.
    const float x0f = fminf(fmaxf(floorf(X), 0.0f), 34.0f);
    const float y0f = fminf(fmaxf(floorf(Y), 0.0f), 34.0f);
    const float z0f = fminf(fmaxf(floorf(Z), 0.0f), 34.0f);
    const float xd = X - x0f, yd = Y - y0f, zd = Z - z0f;

    // Padded index -> original index (pad=2 zero-padding handled by bounds check).
    const int xi0 = (int)x0f - 2;
    const int yi0 = (int)y0f - 2;
    const int zi0 = (int)z0f - 2;

    const float wx[2] = {1.0f - xd, xd};
    const float wy[2] = {1.0f - yd, yd};
    const float wz[2] = {1.0f - zd, zd};

    const float* __restrict__ base = in + (size_t)bp * SLAB;

    float a0 = 0.f, a1 = 0.f, a2 = 0.f, a3 = 0.f,
          a4 = 0.f, a5 = 0.f, a6 = 0.f, a7 = 0.f;

#pragma unroll
    for (int ci = 0; ci < 8; ++ci) {
        const int i = (ci >> 2) & 1;   // x-axis corner (indexes H per reference quirk)
        const int j = (ci >> 1) & 1;   // y-axis corner (indexes W)
        const int k =  ci       & 1;   // z-axis corner (indexes D)
        const int ix = xi0 + i, jy = yi0 + j, kz = zi0 + k;
        const bool ok = ((unsigned)ix < (unsigned)NH) &
                        ((unsigned)jy < (unsigned)NW) &
                        ((unsigned)kz < (unsigned)ND);
        if (ok) {
            const float wgt = wx[i] * wy[j] * wz[k];
            const float4* p =
                reinterpret_cast<const float4*>(base + (size_t)(((ix * NW + jy) * ND + kz) * NC));
            const float4 vlo = p[0];
            const float4 vhi = p[1];
            a0 = fmaf(wgt, vlo.x, a0);
            a1 = fmaf(wgt, vlo.y, a1);
            a2 = fmaf(wgt, vlo.z, a2);
            a3 = fmaf(wgt, vlo.w, a3);
            a4 = fmaf(wgt, vhi.x, a4);
            a5 = fmaf(wgt, vhi.y, a5);
            a6 = fmaf(wgt, vhi.z, a6);
            a7 = fmaf(wgt, vhi.w, a7);
        }
    }

    // Coalesced 32B store per lane (consecutive lanes -> consecutive channels blocks).
    float4* o = reinterpret_cast<float4*>(out + (size_t)vid * NC);
    o[0] = make_float4(a0, a1, a2, a3);
    o[1] = make_float4(a4, a5, a6, a7);
}

extern "C" void kernel_launch(void* const* d_in, const int* in_sizes, int n_in,
                              void* d_out, int out_size, void* d_ws, size_t ws_size,
                              hipStream_t stream) {
    (void)in_sizes; (void)n_in; (void)d_ws; (void)ws_size; (void)out_size;
    const float* in_fmap = (const float*)d_in[0];   // (16,4,32,32,32,8) f32
    const float* theta   = (const float*)d_in[1];   // (16,4,3,4) f32
    float* out           = (float*)d_out;           // (16,4,32,32,32,8) f32

    const int threads = 256;                         // 8 waves (wave32)
    const int blocks  = NVOX / threads;              // 8192, exact cover (no tail)
    Resampling_79353815760907_kernel<<<blocks, threads, 0, stream>>>(in_fmap, theta, out);
}